// LogicGatedSNN_10840497455709
// MI455X (gfx1250) — compile-verified
//
#include <hip/hip_runtime.h>
#include <stdint.h>

// ---------------------------------------------------------------------------
// LogicGatedSNN fused step for MI455X (gfx1250, wave32).
// Bandwidth-bound: 768 MB of mandatory HBM traffic -> ~33us floor @ 23.3 TB/s.
// One workgroup per output row; synapse row streamed via CDNA5 async
// global->LDS copies (ASYNCcnt, issue-2-ahead ring), trace row streamed with
// non-temporal loads/stores. WMMA intentionally unused: N=1 GEMV on a
// runtime-thresholded f32 matrix gains nothing from matrix pipes.
// ---------------------------------------------------------------------------

typedef float v4f __attribute__((ext_vector_type(4)));
typedef int   v4i __attribute__((vector_size(16)));     // matches builtin param
typedef __attribute__((address_space(1))) v4i  gv4i;    // global int4
typedef __attribute__((address_space(3))) v4i  lv4i;    // LDS int4
typedef __attribute__((address_space(1))) void gvoid;
typedef __attribute__((address_space(3))) void lvoid;

#define IN_F   8192
#define OUT_F  8192
#define THRESH 50.0f
#define BLOCK  256
#define WAVES  (BLOCK / 32)        // 8 waves (wave32)
#define WAVE_SEG (IN_F / WAVES)    // 1024 floats per wave
#define CHUNK  128                 // floats per wave per async chunk (32 lanes * 4)
#define NCHUNK (WAVE_SEG / CHUNK)  // 8 chunks per wave
#define NBUF   3                   // ring depth (issue 2 ahead, reuse after 3)

#if __has_builtin(__builtin_amdgcn_global_load_async_to_lds_b128)
#define HAVE_ASYNC_LDS 1
#else
#define HAVE_ASYNC_LDS 0
#endif

extern "C" __global__ void __launch_bounds__(BLOCK)
snn_fused_kernel(const float* __restrict__ x,     // [IN_F] binary spikes
                 const float* __restrict__ syn,   // [OUT_F, IN_F]
                 const float* __restrict__ mp,    // [OUT_F]
                 const float* __restrict__ thr,   // [OUT_F]
                 const float* __restrict__ tr,    // [OUT_F, IN_F]
                 float* __restrict__ out)         // [spikes | v_new | trace]
{
    __shared__ float xlds[IN_F];                  // 32 KB
#if HAVE_ASYNC_LDS
    __shared__ float stage[WAVES][NBUF][CHUNK];   // 12 KB per-wave ring buffers
#endif
    __shared__ float partials[WAVES];
    __shared__ float spike_sh;

    const int tid  = threadIdx.x;
    const int o    = blockIdx.x;
    const int wave = tid >> 5;
    const int lane = tid & 31;

    const float* __restrict__ row = syn + (size_t)o * IN_F;

#if HAVE_ASYNC_LDS
    const int base = wave * WAVE_SEG + lane * 4;
    // Kick off the synapse-row HBM stream immediately (chunks 0 and 1),
    // overlapping with the x staging below. Per-wave buffers: no barrier
    // needed, completion tracked by this wave's ASYNCcnt (in-order).
    __builtin_amdgcn_global_load_async_to_lds_b128(
        (gv4i*)(gvoid*)(row + base),
        (lv4i*)(lvoid*)&stage[wave][0][lane * 4], 0, 0);
    __builtin_amdgcn_global_load_async_to_lds_b128(
        (gv4i*)(gvoid*)(row + base + CHUNK),
        (lv4i*)(lvoid*)&stage[wave][1][lane * 4], 0, 0);
#endif

    // ---- stage x into LDS (32 KB, L2-resident source); fixed trip count ----
    {
        const v4f* __restrict__ x4 = (const v4f*)x;
        v4f* xl4 = (v4f*)xlds;
#pragma unroll
        for (int k = 0; k < IN_F / 4 / BLOCK; ++k)
            xl4[tid + k * BLOCK] = x4[tid + k * BLOCK];
    }
    __syncthreads();

    // ---- Phase 1: current[o] = sum_i x[i] * (syn[o,i] > 50) ----
    float acc = 0.0f;

#if HAVE_ASYNC_LDS
    {
#pragma unroll
        for (int c = 0; c < NCHUNK; ++c) {
            if (c < NCHUNK - 2) {
                __builtin_amdgcn_global_load_async_to_lds_b128(
                    (gv4i*)(gvoid*)(row + base + (c + 2) * CHUNK),
                    (lv4i*)(lvoid*)&stage[wave][(c + 2) % NBUF][lane * 4], 0, 0);
                asm volatile("s_wait_asynccnt 0x2" ::: "memory");
            } else if (c == NCHUNK - 2) {
                asm volatile("s_wait_asynccnt 0x1" ::: "memory");
            } else {
                asm volatile("s_wait_asynccnt 0x0" ::: "memory");
            }
            v4f s  = *(const v4f*)&stage[wave][c % NBUF][lane * 4];
            v4f xv = *(const v4f*)&xlds[base + c * CHUNK];
            acc += (s.x > THRESH) ? xv.x : 0.0f;
            acc += (s.y > THRESH) ? xv.y : 0.0f;
            acc += (s.z > THRESH) ? xv.z : 0.0f;
            acc += (s.w > THRESH) ? xv.w : 0.0f;
        }
    }
#else
    // fallback: direct non-temporal streaming loads
#pragma unroll 4
    for (int i = tid * 4; i < IN_F; i += BLOCK * 4) {
        v4f s  = __builtin_nontemporal_load((const v4f*)(row + i));
        v4f xv = *(const v4f*)&xlds[i];
        acc += (s.x > THRESH) ? xv.x : 0.0f;
        acc += (s.y > THRESH) ? xv.y : 0.0f;
        acc += (s.z > THRESH) ? xv.z : 0.0f;
        acc += (s.w > THRESH) ? xv.w : 0.0f;
    }
#endif

    // ---- reduce: wave32 shuffle tree, then cross-wave via LDS ----
#pragma unroll
    for (int off = 16; off > 0; off >>= 1)
        acc += __shfl_down(acc, off, 32);
    if (lane == 0) partials[wave] = acc;
    __syncthreads();

    if (tid == 0) {
        float cur = 0.0f;
#pragma unroll
        for (int w = 0; w < WAVES; ++w) cur += partials[w];
        float v     = fmaf(mp[o], 0.6f, cur);          // 0.6*v + I
        float spike = (v >= thr[o]) ? 1.0f : 0.0f;
        out[o]         = spike;                         // spikes
        out[OUT_F + o] = v * (1.0f - spike) * 0.3f;     // soft-reset v_new
        spike_sh = spike;
    }
    __syncthreads();
    const float spike = spike_sh;

    // ---- Phase 2: trace_new = clip(0.7*trace + spike*x, 0, 3) ----
    // 512 MB of once-touched traffic: non-temporal both directions.
    const float* __restrict__ trow = tr + (size_t)o * IN_F;
    float* __restrict__ orow = out + 2 * OUT_F + (size_t)o * IN_F;
#pragma unroll 4
    for (int i = tid * 4; i < IN_F; i += BLOCK * 4) {
        v4f t  = __builtin_nontemporal_load((const v4f*)(trow + i));
        v4f xv = *(const v4f*)&xlds[i];
        v4f r;
        r.x = fminf(fmaxf(fmaf(t.x, 0.7f, spike * xv.x), 0.0f), 3.0f);
        r.y = fminf(fmaxf(fmaf(t.y, 0.7f, spike * xv.y), 0.0f), 3.0f);
        r.z = fminf(fmaxf(fmaf(t.z, 0.7f, spike * xv.z), 0.0f), 3.0f);
        r.w = fminf(fmaxf(fmaf(t.w, 0.7f, spike * xv.w), 0.0f), 3.0f);
        __builtin_nontemporal_store(r, (v4f*)(orow + i));
    }
}

extern "C" void kernel_launch(void* const* d_in, const int* in_sizes, int n_in,
                              void* d_out, int out_size, void* d_ws, size_t ws_size,
                              hipStream_t stream) {
    (void)in_sizes; (void)n_in; (void)out_size; (void)d_ws; (void)ws_size;
    const float* x   = (const float*)d_in[0];  // spike_input        [1, 8192]
    const float* syn = (const float*)d_in[1];  // synapse_states     [8192, 8192]
    const float* mp  = (const float*)d_in[2];  // membrane_potential [8192]
    const float* thr = (const float*)d_in[3];  // adaptive_threshold [8192]
    const float* tr  = (const float*)d_in[4];  // eligibility_trace  [8192, 8192]
    float* out = (float*)d_out;                // [8192 | 8192 | 8192*8192]

    snn_fused_kernel<<<OUT_F, BLOCK, 0, stream>>>(x, syn, mp, thr, tr, out);
}